// HyperConnections_31576599560736
// MI455X (gfx1250) — compile-verified
//
#include <hip/hip_runtime.h>
#include <math.h>

typedef __attribute__((ext_vector_type(2))) float v2f;
typedef __attribute__((ext_vector_type(4))) float v4f;
typedef __attribute__((ext_vector_type(8))) float v8f;
typedef __attribute__((ext_vector_type(4))) int   v4i;

#define EPS_F 1.1920929e-07f
#define DDIM 1024
#define NSTREAM 4
#define NTOK 4096               // N (sequence length)
#define CSTR 1036               // cw row stride (floats): 12*t mod 64 banks, conflict-free
#define KCHUNK 128              // d-chunk staged per WMMA round
#define KSTR 132                // chunk row stride: 4*r mod 64 banks, conflict-free, 16B aligned
#define WAVES 8
#define TOK_PER_WAVE 4
#define ROWS 16                 // 4 tokens * 4 streams per wave

// LDS layout (floats):
//   cw   [8][CSTR]              combined weights (5 alpha cols + beta col + 2 zero rows)
//   albuf[WAVES][ROWS][8]       alpha(5)+beta(1) broadcast table
//   chk  [WAVES][ROWS][KSTR]    per-wave K-chunk transpose buffer
#define CW_OFF 0
#define AL_OFF (8 * CSTR)                        // 8288
#define CH_OFF (AL_OFF + WAVES * ROWS * 8)       // 9312
#define SMEM_FLOATS (CH_OFF + WAVES * ROWS * KSTR)  // 26208 floats = 104832 B -> 3 blocks/WGP

#define AS1 __attribute__((address_space(1)))
#define AS3 __attribute__((address_space(3)))

#if defined(__has_builtin)
#if __has_builtin(__builtin_amdgcn_global_load_async_to_lds_b128)
#define USE_ASYNC_LDS 1
#endif
#endif

extern "C" __global__ void __launch_bounds__(256)
hyperconn_wmma_kernel(const float* __restrict__ residuals,
                      const float* __restrict__ norm_weight,
                      const float* __restrict__ static_alpha,   // (4,5)
                      const float* __restrict__ static_beta,    // (4,)
                      const float* __restrict__ w_alpha,        // (1024,5)
                      const float* __restrict__ a_scale_p,
                      const float* __restrict__ w_beta,         // (1024,)
                      const float* __restrict__ b_scale_p,
                      float* __restrict__ out)
{
    extern __shared__ float smem[];
    float* cw    = smem + CW_OFF;
    float* albuf = smem + AL_OFF;

    const int tid  = threadIdx.x;
    const int lane = tid & 31;
    const int wv   = tid >> 5;
    float* chk = smem + CH_OFF + wv * (ROWS * KSTR);

    // ---- stage combined weights: cw[t][d] = norm_weight[d] * W[d][t]; rows 6,7 = 0
    for (int d = tid; d < DDIM; d += 256) {
        const float w = norm_weight[d];
#pragma unroll
        for (int t = 0; t < 5; ++t) cw[t * CSTR + d] = w * w_alpha[d * 5 + t];
        cw[5 * CSTR + d] = w * w_beta[d];
        cw[6 * CSTR + d] = 0.f;
        cw[7 * CSTR + d] = 0.f;
    }

    const float a_scale = a_scale_p[0];
    const float b_scale = b_scale_p[0];

    const int tokBase = (blockIdx.x * WAVES + wv) * TOK_PER_WAVE;

    // per-row global base pointers (row = tokLocal*4 + s)
    const float* rowsrc[ROWS];
#pragma unroll
    for (int row = 0; row < ROWS; ++row) {
        const int tok = tokBase + (row >> 2);
        const int s   = row & 3;
        const int b   = tok / NTOK;
        const int n   = tok - b * NTOK;
        rowsrc[row] = residuals + ((size_t)(b * NSTREAM + s) * NTOK + n) * DDIM;
    }

    __syncthreads();   // cw visible to all waves

    // ---- Phase A+B fused: for each 128-wide K chunk, stage 16 rows to LDS
    // (async DMA -> LDS), then 32 x V_WMMA_F32_16X16X4_F32.
    // A (16x4 f32): lane<16 -> (M=lane, K=0..1), lane>=16 -> (M=lane-16, K=2..3) => ds_load_b64
    // B (4x16 f32): lanes 0-15 hold K=0..1, lanes 16-31 hold K=2..3, N=lane%16   => ds_load_b64
    const int hi = lane >> 4;
    const int lm = lane & 15;
    const int tcl = (lm < 8) ? lm : 7;           // cols >= 6 hit zero rows
    const float* arow_base = chk + lm * KSTR + 2 * hi;
    const float* brow_base = cw + tcl * CSTR + 2 * hi;

    v8f acc = {0.f, 0.f, 0.f, 0.f, 0.f, 0.f, 0.f, 0.f};
    float sp = 0.f;    // per-lane sumsq partial; lane L accumulates only row (L&15)

    for (int c = 0; c < DDIM / KCHUNK; ++c) {
        // stage chunk c: one 512B-coalesced transfer per row
#if defined(USE_ASYNC_LDS)
#pragma unroll
        for (int row = 0; row < ROWS; ++row) {
            const float* src = rowsrc[row] + c * KCHUNK + lane * 4;
            float* dst = chk + row * KSTR + lane * 4;
            __builtin_amdgcn_global_load_async_to_lds_b128((AS1 v4i*)src, (AS3 v4i*)dst,
                                                           0, 0);
        }
        __asm__ volatile("s_wait_asynccnt 0x0" ::: "memory");
#else
#pragma unroll
        for (int row = 0; row < ROWS; ++row) {
            v4f v = *reinterpret_cast<const v4f*>(rowsrc[row] + c * KCHUNK + lane * 4);
            *reinterpret_cast<v4f*>(chk + row * KSTR + lane * 4) = v;
        }
        __asm__ volatile("s_wait_dscnt 0x0" ::: "memory");
#endif
        const float* arow = arow_base;
        const float* brow = brow_base + c * KCHUNK;
#pragma unroll
        for (int cc = 0; cc < KCHUNK / 4; ++cc) {
            v2f a = *reinterpret_cast<const v2f*>(arow + 4 * cc);
            v2f b = *reinterpret_cast<const v2f*>(brow + 4 * cc);
            sp += a.x * a.x + a.y * a.y;        // raw-r sumsq rides on the A fragments
            acc = __builtin_amdgcn_wmma_f32_16x16x4_f32(false, a, false, b, (short)0, acc,
                                                        false, false);
        }
    }

    // row sumsq: lanes m and m+16 jointly hold row m
    float tot = sp + __shfl_xor(sp, 16, 32);
    const float rstd_row = rsqrtf(tot * (1.0f / DDIM) + EPS_F);   // for row (lane&15)

    // fold rstd post-GEMM (acc[j] is row j+8*hi) and apply nonlinearity + static offsets
    float logit[8];
#pragma unroll
    for (int j = 0; j < 8; ++j)
        logit[j] = acc[j] * __shfl(rstd_row, j + 8 * hi, 32);

    if (lm < 5) {
#pragma unroll
        for (int j = 0; j < 8; ++j) {
            const int row = j + 8 * hi;
            albuf[(wv * ROWS + row) * 8 + lm] =
                tanhf(logit[j]) * a_scale + static_alpha[(row & 3) * 5 + lm];
        }
    } else if (lm == 5) {
#pragma unroll
        for (int j = 0; j < 8; ++j) {
            const int row = j + 8 * hi;
            albuf[(wv * ROWS + row) * 8 + 5] =
                tanhf(logit[j]) * b_scale + static_beta[row & 3];
        }
    }
    __asm__ volatile("s_wait_dscnt 0x0" ::: "memory");   // albuf visible within wave

    // ---- Phase C: mix + beta residual; re-read r from global (L2-hot), stream out (b128)
#pragma unroll
    for (int tl = 0; tl < TOK_PER_WAVE; ++tl) {
        float al[NSTREAM][5], bt[NSTREAM];
#pragma unroll
        for (int s = 0; s < NSTREAM; ++s) {
            const int row = tl * 4 + s;
#pragma unroll
            for (int t = 0; t < 5; ++t) al[s][t] = albuf[(wv * ROWS + row) * 8 + t];
            bt[s] = albuf[(wv * ROWS + row) * 8 + 5];
        }
        const int tok = tokBase + tl;
        const int b   = tok / NTOK;
        const int n   = tok - b * NTOK;
        const float* r0 = rowsrc[tl * 4 + 0];
        const float* r1 = rowsrc[tl * 4 + 1];
        const float* r2 = rowsrc[tl * 4 + 2];
        const float* r3 = rowsrc[tl * 4 + 3];
        float* o0 = out + ((size_t)(b * NSTREAM + 0) * NTOK + n) * DDIM;
        float* o1 = out + ((size_t)(b * NSTREAM + 1) * NTOK + n) * DDIM;
        float* o2 = out + ((size_t)(b * NSTREAM + 2) * NTOK + n) * DDIM;
        float* o3 = out + ((size_t)(b * NSTREAM + 3) * NTOK + n) * DDIM;
#pragma unroll
        for (int j = 0; j < 8; ++j) {
            const int d = j * 128 + lane * 4;
            v4f x0 = *reinterpret_cast<const v4f*>(r0 + d);
            v4f x1 = *reinterpret_cast<const v4f*>(r1 + d);
            v4f x2 = *reinterpret_cast<const v4f*>(r2 + d);
            v4f x3 = *reinterpret_cast<const v4f*>(r3 + d);
            v4f m0 = al[0][0] * x0 + al[1][0] * x1 + al[2][0] * x2 + al[3][0] * x3;
            v4f y0 = bt[0] * m0 + al[0][1] * x0 + al[1][1] * x1 + al[2][1] * x2 + al[3][1] * x3;
            v4f y1 = bt[1] * m0 + al[0][2] * x0 + al[1][2] * x1 + al[2][2] * x2 + al[3][2] * x3;
            v4f y2 = bt[2] * m0 + al[0][3] * x0 + al[1][3] * x1 + al[2][3] * x2 + al[3][3] * x3;
            v4f y3 = bt[3] * m0 + al[0][4] * x0 + al[1][4] * x1 + al[2][4] * x2 + al[3][4] * x3;
            *reinterpret_cast<v4f*>(o0 + d) = y0;
            *reinterpret_cast<v4f*>(o1 + d) = y1;
            *reinterpret_cast<v4f*>(o2 + d) = y2;
            *reinterpret_cast<v4f*>(o3 + d) = y3;
        }
    }
}

extern "C" void kernel_launch(void* const* d_in, const int* in_sizes, int n_in,
                              void* d_out, int out_size, void* d_ws, size_t ws_size,
                              hipStream_t stream) {
    const float* residuals    = (const float*)d_in[0];
    const float* norm_weight  = (const float*)d_in[1];
    const float* static_alpha = (const float*)d_in[2];
    const float* static_beta  = (const float*)d_in[3];
    const float* w_alpha      = (const float*)d_in[4];
    const float* a_scale      = (const float*)d_in[5];
    const float* w_beta       = (const float*)d_in[6];
    const float* b_scale      = (const float*)d_in[7];
    float* out = (float*)d_out;

    const int D = in_sizes[1];                       // 1024
    const int S = in_sizes[3];                       // 4
    const int totalTokens = in_sizes[0] / (S * D);   // B*N = 16384
    const int blocks = totalTokens / (WAVES * TOK_PER_WAVE);   // 512

    const int smem_bytes = SMEM_FLOATS * (int)sizeof(float);   // ~102 KB -> 3 blocks/WGP
    (void)hipFuncSetAttribute(reinterpret_cast<const void*>(hyperconn_wmma_kernel),
                              hipFuncAttributeMaxDynamicSharedMemorySize, smem_bytes);

    hyperconn_wmma_kernel<<<blocks, 256, smem_bytes, stream>>>(
        residuals, norm_weight, static_alpha, static_beta,
        w_alpha, a_scale, w_beta, b_scale, out);
}